// SimilarityBasedRelationEnhancer_74251394613539
// MI455X (gfx1250) — compile-verified
//
#include <hip/hip_runtime.h>
#include <hip/hip_bf16.h>
#include <math.h>

// Problem constants from the reference (D and K are fixed by the model).
#define DIM 64
#define KTOP 20
#define TILE_ROWS 128          // rows per workgroup in the streaming kernel
#define ROWSTRIDE 68           // padded LDS row stride (floats): conflict-free 8B reads

typedef __attribute__((ext_vector_type(2))) float v2f;
typedef __attribute__((ext_vector_type(4))) float v4f;   // native vector: OK for NT builtins
typedef __attribute__((ext_vector_type(8))) float v8f;

#if defined(__has_builtin)
#if __has_builtin(__builtin_amdgcn_wmma_f32_16x16x4_f32)
#define HAS_WMMA4 1
#endif
#endif
#ifndef HAS_WMMA4
#define HAS_WMMA4 0
#endif

// ---------------------------------------------------------------------------
// Kernel 1: per-batch, normalize the query row into ws.qn  (1 wave / batch)
// ---------------------------------------------------------------------------
__global__ void prep_q_kernel(const float* __restrict__ x,
                              const int* __restrict__ qrel,
                              float* __restrict__ qn, int R) {
  const int b = blockIdx.x;
  const int lane = threadIdx.x;               // 0..31
  const int qr = qrel[b];
  const float2* q = (const float2*)(x + ((size_t)b * R + qr) * DIM);
  float2 v = q[lane];
  float s = v.x * v.x + v.y * v.y;
  #pragma unroll
  for (int off = 16; off >= 1; off >>= 1) s += __shfl_xor(s, off);
  const float inv = 1.0f / fmaxf(sqrtf(s), 1e-12f);
  float2 o; o.x = v.x * inv; o.y = v.y * inv;
  ((float2*)(qn + (size_t)b * DIM))[lane] = o;
}

// ---------------------------------------------------------------------------
// Kernel 2: fused stream: copy x->out, and per row compute sims & 1/norm.
// Grid: (R/TILE_ROWS, B), 256 threads (8 waves x 16 rows each).
// Dots via V_WMMA_F32_16X16X4_F32: A = 16 rows x 4 dims, B = q-chunk
// broadcast to all 16 columns -> D[m,*] = dot(row m, qn).
// 'out' stores are non-temporal: out is never re-read, and keeping it out of
// L2 lets the 128MB x tensor stay L2-resident across graph replays
// (192MB L2), so steady-state HBM traffic ~= the write stream only.
// ---------------------------------------------------------------------------
__global__ void sims_copy_kernel(const float* __restrict__ x,
                                 const float* __restrict__ qn,
                                 float* __restrict__ out,
                                 float* __restrict__ sims,
                                 float* __restrict__ rinv, int R) {
  __shared__ float tile[TILE_ROWS * ROWSTRIDE];
  __shared__ float qs[DIM];

  const int b    = blockIdx.y;
  const int row0 = blockIdx.x * TILE_ROWS;
  const int t    = threadIdx.x;                 // 0..255
  const size_t base = ((size_t)b * R + row0) * DIM;

  if (t < DIM) qs[t] = qn[(size_t)b * DIM + t];

  // --- Phase 1: stream tile (coalesced 16B): global -> out + LDS ---
  const v4f* src = (const v4f*)(x + base);
  v4f*       dst = (v4f*)(out + base);
  #pragma unroll
  for (int i = 0; i < (TILE_ROWS * DIM / 4) / 256; ++i) {
    const int idx = t + i * 256;                // 16B-vector index in tile
    __builtin_prefetch(src + idx + 512, 0, 0);  // global_prefetch_b8
    v4f v = src[idx];
    __builtin_nontemporal_store(v, &dst[idx]);  // TH=NT: don't pollute L2
    const int r = idx >> 4;                     // 16 vec4 per row
    const int c = (idx & 15) << 2;
    *(v4f*)&tile[r * ROWSTRIDE + c] = v;
  }
  __syncthreads();

  // --- Phase 2: per-wave 16-row dot products + norms ---
  const int wave   = t >> 5;
  const int lane   = t & 31;
  const int m      = lane & 15;                 // row within wave's 16
  const int h      = lane >> 4;                 // half: dims {4c,4c+1} vs {4c+2,4c+3}
  const int rlocal = wave * 16;
  const float* arow = &tile[(rlocal + m) * ROWSTRIDE];

  float nsq = 0.0f;
#if HAS_WMMA4
  v8f acc = {};
  #pragma unroll
  for (int c = 0; c < 16; ++c) {
    const int col = 4 * c + 2 * h;
    v2f a;  a.x  = arow[col]; a.y  = arow[col + 1];
    v2f bb; bb.x = qs[col];   bb.y = qs[col + 1];
    acc = __builtin_amdgcn_wmma_f32_16x16x4_f32(false, a, false, bb,
                                                (short)0, acc, false, false);
    nsq += a.x * a.x + a.y * a.y;
  }
  nsq += __shfl_xor(nsq, 16);
  const float ri = 1.0f / fmaxf(sqrtf(nsq), 1e-12f);
  if (lane < 16) rinv[(size_t)b * R + row0 + rlocal + m] = ri;
  // D layout: vgpr v on lane L holds row (v + 8*(L>>4)), any column.
  // lanes 0 and 16 emit the 16 sims; shfl gathers the matching 1/norm.
  #pragma unroll
  for (int v = 0; v < 8; ++v) {
    const float rv = __shfl(ri, (h << 3) + v);  // rinv of row (h*8 + v)
    if (m == 0) {
      const int row = row0 + rlocal + (h << 3) + v;
      sims[(size_t)b * R + row] = acc[v] * rv;
    }
  }
#else
  // VALU fallback (q already unit-norm, so dot * rinv = cosine)
  float dt = 0.0f;
  #pragma unroll
  for (int c = 0; c < 16; ++c) {
    const int col = 4 * c + 2 * h;
    const float a0 = arow[col], a1 = arow[col + 1];
    dt  += a0 * qs[col] + a1 * qs[col + 1];
    nsq += a0 * a0 + a1 * a1;
  }
  dt  += __shfl_xor(dt, 16);
  nsq += __shfl_xor(nsq, 16);
  const float ri = 1.0f / fmaxf(sqrtf(nsq), 1e-12f);
  if (lane < 16) {
    rinv[(size_t)b * R + row0 + rlocal + m] = ri;
    sims[(size_t)b * R + row0 + rlocal + m] = dt * ri;
  }
#endif
}

// ---------------------------------------------------------------------------
// Kernel 3: per-batch top-K + weighting + mix; overwrites out[b, qr, :].
// 256 threads / batch; dynamic LDS holds the sims row.
// ---------------------------------------------------------------------------
__global__ void finalize_kernel(const float* __restrict__ x,
                                const int* __restrict__ qrel,
                                const float* __restrict__ sims,
                                const float* __restrict__ rinv,
                                const float* __restrict__ thr_raw,
                                const float* __restrict__ str_raw,
                                const float* __restrict__ wscale,
                                const float* __restrict__ temper,
                                float* __restrict__ out, int R) {
  extern __shared__ float ls[];                 // R floats
  __shared__ float red_v[256];
  __shared__ int   red_i[256];
  __shared__ float topv[KTOP];
  __shared__ int   topi[KTOP];
  __shared__ float adj[KTOP];
  __shared__ float sc[8];                       // thr, strength, temp, wscale, has_valid

  const int b = blockIdx.x;
  const int t = threadIdx.x;
  const int qr = qrel[b];

  if (t == 0) {
    sc[0] = 1.0f / (1.0f + expf(-thr_raw[0]));          // threshold
    sc[1] = (1.0f / (1.0f + expf(-str_raw[0]))) * 0.2f; // strength
    sc[2] = fminf(fmaxf(temper[0], 0.1f), 10.0f);       // temperature
    sc[3] = wscale[0];
  }
  for (int i = t; i < R; i += 256) ls[i] = sims[(size_t)b * R + i];
  __syncthreads();
  if (t == 0) ls[qr] = -1.0f;                   // exclude self (matches reference)
  __syncthreads();

  // exact top-K, tie-break = lowest index (matches lax.top_k)
  const int per = R / 256;
  for (int k = 0; k < KTOP; ++k) {
    float bv = -1e30f; int bi = 0x7fffffff;
    for (int j = 0; j < per; ++j) {
      const int i = t * per + j;                // increasing index -> first win on tie
      const float v = ls[i];
      if (v > bv) { bv = v; bi = i; }
    }
    red_v[t] = bv; red_i[t] = bi;
    __syncthreads();
    for (int off = 128; off > 0; off >>= 1) {
      if (t < off) {
        const float v2 = red_v[t + off]; const int i2 = red_i[t + off];
        if (v2 > red_v[t] || (v2 == red_v[t] && i2 < red_i[t])) {
          red_v[t] = v2; red_i[t] = i2;
        }
      }
      __syncthreads();
    }
    if (t == 0) {
      topv[k] = red_v[0]; topi[k] = red_i[0];
      ls[red_i[0]] = -1e30f;
    }
    __syncthreads();
  }

  if (t == 0) {
    const float thr = sc[0], tmp = sc[2], wsc = sc[3];
    float logits[KTOP]; bool valid[KTOP];
    float mx = -1e30f; int any = 0;
    for (int k = 0; k < KTOP; ++k) {
      const bool va = (topv[k] > thr) && (topi[k] != qr);
      valid[k] = va; any |= (int)va;
      logits[k] = va ? topv[k] / tmp : -1e9f;
      mx = fmaxf(mx, logits[k]);
    }
    float se = 0.0f; float e[KTOP];
    for (int k = 0; k < KTOP; ++k) { e[k] = expf(logits[k] - mx); se += e[k]; }
    float asum = 0.0f;
    for (int k = 0; k < KTOP; ++k) {
      const float w  = valid[k] ? (e[k] / se) : 0.0f;
      const float sw = 1.0f / (1.0f + expf(-(topv[k] - thr) * 10.0f));
      const float a  = w * sw * (1.0f + wsc * topv[k]);
      adj[k] = a; asum += a;
    }
    const float inv = 1.0f / (asum + 1e-8f);
    for (int k = 0; k < KTOP; ++k) adj[k] *= inv;
    sc[4] = any ? 1.0f : 0.0f;
  }
  __syncthreads();

  if (t < DIM) {
    float acc = 0.0f;
    for (int k = 0; k < KTOP; ++k) {
      const int ti = topi[k];
      acc += adj[k] * x[((size_t)b * R + ti) * DIM + t] * rinv[(size_t)b * R + ti];
    }
    const float qo  = x[((size_t)b * R + qr) * DIM + t];
    const float str = sc[1];
    const float mixed = (1.0f - str) * qo + str * acc;
    out[((size_t)b * R + qr) * DIM + t] = (sc[4] != 0.0f) ? mixed : qo;
  }
}

// ---------------------------------------------------------------------------
extern "C" void kernel_launch(void* const* d_in, const int* in_sizes, int n_in,
                              void* d_out, int out_size, void* d_ws, size_t ws_size,
                              hipStream_t stream) {
  const float* x       = (const float*)d_in[0];
  const int*   qrel    = (const int*)d_in[1];
  const float* thr_raw = (const float*)d_in[2];
  const float* str_raw = (const float*)d_in[3];
  const float* wscale  = (const float*)d_in[4];
  const float* temper  = (const float*)d_in[5];
  float* out = (float*)d_out;

  const int B = in_sizes[1];
  const int R = in_sizes[0] / (B * DIM);

  // workspace layout
  float* qn   = (float*)d_ws;                 // B * DIM
  float* sims = qn + (size_t)B * DIM;         // B * R
  float* rinv = sims + (size_t)B * R;         // B * R

  prep_q_kernel<<<B, 32, 0, stream>>>(x, qrel, qn, R);

  dim3 g2(R / TILE_ROWS, B);
  sims_copy_kernel<<<g2, 256, 0, stream>>>(x, qn, out, sims, rinv, R);

  finalize_kernel<<<B, 256, (size_t)R * sizeof(float), stream>>>(
      x, qrel, sims, rinv, thr_raw, str_raw, wscale, temper, out, R);
}